// Model_16569983828187
// MI455X (gfx1250) — compile-verified
//
#include <hip/hip_runtime.h>
#include <hip/hip_bf16.h>

// ---------------------------------------------------------------------------
// NMS for MI455X (gfx1250, wave32).
// Stage 1: build 64-bit stable-sort keys.
// Stage 2: O(N^2) rank-and-scatter sort with LDS-tiled keys (async global->LDS
//          loads on the ASYNCcnt path: global_load_async_to_lds_b64 +
//          s_wait_asynccnt — the CDNA5-specific data-movement path).
// Stage 3: single-workgroup sequential greedy NMS with the suppression bitmask
//          held in LDS (word-per-thread suppression, no atomics, deterministic).
// ---------------------------------------------------------------------------

#define SORT_BLOCK 256
#define KEY_TILE   2048           // 16 KB of LDS per block for key tiles
#define MAX_WORDS  1024           // supports up to 32768 boxes (N = 20000 here)

#if defined(__has_builtin)
#  if __has_builtin(__builtin_amdgcn_global_load_async_to_lds_b64)
#    define NMS_ASYNC_LDS 1
#  endif
#endif

#if defined(NMS_ASYNC_LDS)
typedef int v2i __attribute__((vector_size(8)));   // matches builtin param type
typedef __attribute__((address_space(1))) v2i* v2i_global_ptr;
typedef __attribute__((address_space(3))) v2i* v2i_lds_ptr;
// AMDGPU flat addresses: global flat address == global address; LDS flat
// address keeps the byte offset in bits [31:0] (ISA 10.2 aperture rules),
// so integer round-trips give us correctly typed addrspace pointers.
#define AS_GLOBAL_V2I(p) ((v2i_global_ptr)(unsigned long long)(p))
#define AS_LDS_V2I(p)    ((v2i_lds_ptr)(unsigned)(unsigned long long)(p))
#endif

// key = (orderable(score) << 32) | (0xFFFFFFFF - i)
// Descending sort on this key == stable descending sort by score with
// ties broken by ascending original index (matches jnp.argsort(-key)).
__global__ void nms_build_keys(const float* __restrict__ scores,
                               const float* __restrict__ sthr_p, int n,
                               unsigned long long* __restrict__ keys) {
  int i = blockIdx.x * blockDim.x + threadIdx.x;
  if (i >= n) return;
  const float thr = *sthr_p;
  float s = scores[i];
  float k = (s > thr) ? s : __uint_as_float(0xFF800000u);   // -inf for invalid
  unsigned u = __float_as_uint(k);
  u = (u & 0x80000000u) ? ~u : (u | 0x80000000u);           // monotone map
  keys[i] = ((unsigned long long)u << 32) | (unsigned)(0xFFFFFFFFu - (unsigned)i);
}

// rank[i] = #{ j : key[j] > key[i] }  (keys are all distinct) -> unique rank.
// Scatter box i to sorted position rank[i].
__global__ void nms_rank_scatter(const unsigned long long* __restrict__ keys,
                                 const float* __restrict__ boxes,
                                 const float* __restrict__ scores,
                                 const float* __restrict__ sthr_p, int n,
                                 float4* __restrict__ sboxes,
                                 float* __restrict__ sarea,
                                 int*   __restrict__ sorig,
                                 unsigned char* __restrict__ ssup) {
  __shared__ unsigned long long tile[KEY_TILE];
  const int i = blockIdx.x * blockDim.x + threadIdx.x;
  const unsigned long long myKey = (i < n) ? keys[i] : 0ull;
  int rank = 0;
  for (int base = 0; base < n; base += KEY_TILE) {
    const int cnt = min(KEY_TILE, n - base);
#if defined(NMS_ASYNC_LDS)
    for (int t = threadIdx.x; t < cnt; t += blockDim.x)
      __builtin_amdgcn_global_load_async_to_lds_b64(AS_GLOBAL_V2I(keys + base + t),
                                                    AS_LDS_V2I(&tile[t]), 0, 0);
#  if defined(__has_builtin) && __has_builtin(__builtin_amdgcn_s_wait_asynccnt)
    __builtin_amdgcn_s_wait_asynccnt(0);
#  else
    asm volatile("s_wait_asynccnt 0" ::: "memory");
#  endif
#else
    for (int t = threadIdx.x; t < cnt; t += blockDim.x) tile[t] = keys[base + t];
#endif
    __syncthreads();
    if (i < n) {
      for (int t = 0; t < cnt; ++t) rank += (tile[t] > myKey) ? 1 : 0;
    }
    __syncthreads();
  }
  if (i < n) {
    const float x1 = boxes[4 * i + 0], y1 = boxes[4 * i + 1];
    const float x2 = boxes[4 * i + 2], y2 = boxes[4 * i + 3];
    sboxes[rank] = make_float4(x1, y1, x2, y2);
    sarea[rank]  = (x2 - x1) * (y2 - y1);
    sorig[rank]  = i;
    ssup[rank]   = (scores[i] > *sthr_p) ? 0 : 1;   // invalid => pre-suppressed
  }
}

// Sequential greedy NMS: one workgroup (32 wave32 waves).
__global__ void __launch_bounds__(1024)
nms_sequential(const float4* __restrict__ sboxes,
               const float* __restrict__ sarea,
               const int*   __restrict__ sorig,
               const unsigned char* __restrict__ ssup, int n,
               const int*   __restrict__ maxout_p,
               const float* __restrict__ iou_p,
               int* __restrict__ out) {
  __shared__ unsigned sup[MAX_WORDS];
  __shared__ int s_cnt;
  __shared__ int s_sel;
  __shared__ float s_box[5];

  const int   tid     = threadIdx.x;
  const int   nt      = blockDim.x;
  const int   max_out = *maxout_p;
  const float thr     = *iou_p;
  const int   words   = (n + 31) >> 5;

  // zero the selected-index output (reference initializes with zeros)
  for (int k = tid; k < max_out; k += nt) out[k] = 0;

  // build suppression bitmask from validity; bits past n are marked suppressed
  for (int w = tid; w < words; w += nt) {
    const int base = w << 5;
    const int lim  = min(32, n - base);
    unsigned m = (lim < 32) ? (0xFFFFFFFFu << lim) : 0u;
    for (int b = 0; b < lim; ++b) m |= (unsigned)(ssup[base + b] & 1) << b;
    sup[w] = m;
  }
  if (tid == 0) s_cnt = 0;
  __syncthreads();

  int scan_i = 0;  // only meaningful on thread 0
  for (;;) {
    if (tid == 0) {
      int sel = -1;
      if (s_cnt < max_out) {
        int i = scan_i;
        while (i < n) {
          const unsigned avail = ~sup[i >> 5] & (0xFFFFFFFFu << (i & 31));
          if (avail) { i = (i & ~31) + __builtin_ctz(avail); break; }
          i = (i & ~31) + 32;
        }
        if (i < n) {
          sel = i;
          scan_i = i + 1;
          out[s_cnt] = sorig[i];
          s_cnt = s_cnt + 1;
          const float4 b = sboxes[i];
          s_box[0] = b.x; s_box[1] = b.y; s_box[2] = b.z; s_box[3] = b.w;
          s_box[4] = sarea[i];
        }
      }
      s_sel = sel;
    }
    __syncthreads();
    const int sel = s_sel;
    if (sel < 0) break;                     // uniform exit

    const float bx1 = s_box[0], by1 = s_box[1];
    const float bx2 = s_box[2], by2 = s_box[3];
    const float ba  = s_box[4];
    // word-per-thread suppression of all j > sel (each word owned by one thread)
    for (int w = ((sel + 1) >> 5) + tid; w < words; w += nt) {
      const int base = w << 5;
      const int jlo  = max(base, sel + 1);
      const int jhi  = min(base + 32, n);
      unsigned add = 0;
      for (int j = jlo; j < jhi; ++j) {
        const float4 c   = sboxes[j];
        const float xx1  = fmaxf(bx1, c.x);
        const float yy1  = fmaxf(by1, c.y);
        const float xx2  = fminf(bx2, c.z);
        const float yy2  = fminf(by2, c.w);
        const float inter = fmaxf(xx2 - xx1, 0.0f) * fmaxf(yy2 - yy1, 0.0f);
        const float uni   = fmaxf(ba + sarea[j] - inter, 1e-6f);
        if (inter / uni >= thr) add |= 1u << (j - base);
      }
      if (add) sup[w] |= add;
    }
    __syncthreads();                        // publish suppression updates
  }

  __syncthreads();
  if (tid == 0) out[max_out] = s_cnt;       // count goes right after selected[]
}

extern "C" void kernel_launch(void* const* d_in, const int* in_sizes, int n_in,
                              void* d_out, int out_size, void* d_ws, size_t ws_size,
                              hipStream_t stream) {
  (void)n_in; (void)out_size; (void)ws_size;

  const float* boxes   = (const float*)d_in[0];
  const float* scores  = (const float*)d_in[1];
  const int*   d_max   = (const int*)d_in[2];     // device scalar
  const float* d_iou   = (const float*)d_in[3];   // device scalar
  const float* d_sthr  = (const float*)d_in[4];   // device scalar
  int*         out     = (int*)d_out;             // [max_out] indices, then count
  const int    n       = in_sizes[1];

  // carve workspace (≈ 33*N bytes ≈ 660 KB for N=20000)
  char* ws = (char*)d_ws;
  size_t off = 0;
  auto carve = [&](size_t bytes) -> char* {
    char* p = ws + off;
    off = (off + bytes + 255) & ~(size_t)255;
    return p;
  };
  unsigned long long* keys  = (unsigned long long*)carve((size_t)n * 8);
  float4*             sbox  = (float4*)carve((size_t)n * 16);
  float*              sarea = (float*)carve((size_t)n * 4);
  int*                sorig = (int*)carve((size_t)n * 4);
  unsigned char*      ssup  = (unsigned char*)carve((size_t)n);

  const dim3 blk(SORT_BLOCK);
  const dim3 grd((n + SORT_BLOCK - 1) / SORT_BLOCK);

  nms_build_keys<<<grd, blk, 0, stream>>>(scores, d_sthr, n, keys);
  nms_rank_scatter<<<grd, blk, 0, stream>>>(keys, boxes, scores, d_sthr, n,
                                            sbox, sarea, sorig, ssup);
  nms_sequential<<<1, 1024, 0, stream>>>(sbox, sarea, sorig, ssup, n,
                                         d_max, d_iou, out);
}